// CodeBook_88115549044791
// MI455X (gfx1250) — compile-verified
//
#include <hip/hip_runtime.h>
#include <math.h>

typedef __attribute__((ext_vector_type(2))) float v2f;
typedef __attribute__((ext_vector_type(8))) float v8f;
typedef __attribute__((ext_vector_type(4))) int   i4v;

#define NTOK  16384   // bs*dim/64
#define BOOK  16384
#define DIM   1024
#define BS    1024
#define CHUNK 128     // codebook rows staged in LDS per iteration

// ---------------------------------------------------------------------------
// f32 WMMA helper: D = A(16x4) * B(4x16) + C(16x16), wave32
// ---------------------------------------------------------------------------
static __device__ inline v8f wmma_f32_4(v2f a, v2f b, v8f c) {
  return __builtin_amdgcn_wmma_f32_16x16x4_f32(
      /*neg_a=*/false, a, /*neg_b=*/false, b,
      /*c_mod=*/(short)0, c, /*reuse_a=*/false, /*reuse_b=*/false);
}

// ---------------------------------------------------------------------------
// gfx1250 async global->LDS copies (ASYNCcnt path) with sync fallback
// ---------------------------------------------------------------------------
#if __has_builtin(__builtin_amdgcn_global_load_async_to_lds_b128) && \
    __has_builtin(__builtin_amdgcn_global_load_async_to_lds_b32)
#define HAS_ASYNC_LDS 1
#else
#define HAS_ASYNC_LDS 0
#endif

#define AS_GLOBAL __attribute__((address_space(1)))
#define AS_LDS    __attribute__((address_space(3)))

static __device__ inline void copy_f4_to_lds(const float4* g, float4* l) {
#if HAS_ASYNC_LDS
  __builtin_amdgcn_global_load_async_to_lds_b128(
      (i4v AS_GLOBAL*)(uintptr_t)g, (i4v AS_LDS*)(uint32_t)(uintptr_t)l, 0, 0);
#else
  *l = *g;
#endif
}
static __device__ inline void copy_f1_to_lds(const float* g, float* l) {
#if HAS_ASYNC_LDS
  __builtin_amdgcn_global_load_async_to_lds_b32(
      (int AS_GLOBAL*)(uintptr_t)g, (int AS_LDS*)(uint32_t)(uintptr_t)l, 0, 0);
#else
  *l = *g;
#endif
}
static __device__ inline void async_wait() {
#if HAS_ASYNC_LDS
  asm volatile("s_wait_asynccnt 0" ::: "memory");
#endif
}

// ---------------------------------------------------------------------------
// Normalize cls sub-tokens: one wave per 64-elem row (2 floats/lane, wave32)
// row i of cls.reshape(-1,64) lives at tokens[(i>>4)*2048 + (i&15)*64 + j]
// ---------------------------------------------------------------------------
__global__ __launch_bounds__(256) void norm_tokens_kernel(
    const float* __restrict__ tokens, float* __restrict__ tn) {
  const int wave = threadIdx.x >> 5, lane = threadIdx.x & 31;
  const int row = blockIdx.x * 8 + wave;
  const float* s = tokens + (size_t)(row >> 4) * 2048 + (size_t)(row & 15) * 64;
  float x0 = s[lane], x1 = s[lane + 32];
  float ss = x0 * x0 + x1 * x1;
#pragma unroll
  for (int m = 16; m >= 1; m >>= 1) ss += __shfl_xor(ss, m, 32);
  float inv = 1.0f / fmaxf(sqrtf(ss), 1e-12f);
  tn[(size_t)row * 64 + lane]      = x0 * inv;
  tn[(size_t)row * 64 + lane + 32] = x1 * inv;
}

// ---------------------------------------------------------------------------
// Normalize codebook rows + emit ||c_n||^2 per code
// ---------------------------------------------------------------------------
__global__ __launch_bounds__(256) void norm_codebook_kernel(
    const float* __restrict__ cb, float* __restrict__ cn,
    float* __restrict__ cn2) {
  const int wave = threadIdx.x >> 5, lane = threadIdx.x & 31;
  const int row = blockIdx.x * 8 + wave;
  const float* s = cb + (size_t)row * 64;
  float x0 = s[lane], x1 = s[lane + 32];
  float ss = x0 * x0 + x1 * x1;
#pragma unroll
  for (int m = 16; m >= 1; m >>= 1) ss += __shfl_xor(ss, m, 32);
  float d = fmaxf(sqrtf(ss), 1e-12f);
  float inv = 1.0f / d;
  cn[(size_t)row * 64 + lane]      = x0 * inv;
  cn[(size_t)row * 64 + lane + 32] = x1 * inv;
  if (lane == 0) cn2[row] = ss * inv * inv;   // == sum(c_n^2)
}

// ---------------------------------------------------------------------------
// Nearest-neighbor search: rows(16/wave) x codes GEMM (K=64) via f32 WMMA,
// fused argmin of dist = ||c_n||^2 - 2*dot  (token norm is a row constant).
// 8 waves / WG, 128 rows / WG; codebook double-buffered through LDS with
// async global->LDS copies overlapping the WMMA chains.
// ---------------------------------------------------------------------------
static __device__ inline void stage_codes(const float* __restrict__ cn,
                                          const float* __restrict__ cn2,
                                          int cb, float* bufC, float* bufN,
                                          int tid) {
  const float4* src = (const float4*)(cn + (size_t)cb * 64);
  float4* dst = (float4*)bufC;
#pragma unroll
  for (int i = 0; i < (CHUNK * 16) / 256; ++i)
    copy_f4_to_lds(src + tid + i * 256, dst + tid + i * 256);
  if (tid < CHUNK) copy_f1_to_lds(cn2 + cb + tid, bufN + tid);
}

__global__ __launch_bounds__(256) void nn_search_kernel(
    const float* __restrict__ tn, const float* __restrict__ cn,
    const float* __restrict__ cn2, int* __restrict__ idx_out) {
  __shared__ float ldsC[2][CHUNK * 64];
  __shared__ float ldsN[2][CHUNK];
  const int tid  = threadIdx.x;
  const int wave = tid >> 5;
  const int lane = tid & 31;
  const int l16  = lane & 15;
  const int half = lane >> 4;
  const int rowbase = blockIdx.x * 128 + wave * 16;

  // Resident A operand: 16 k-steps of (k, k+1) pairs, k = 4*step + 2*half
  v2f a[16];
  {
    const float* ar = tn + (size_t)(rowbase + l16) * 64;
#pragma unroll
    for (int k = 0; k < 16; ++k) {
      const int kk = k * 4 + half * 2;
      a[k].x = ar[kk];
      a[k].y = ar[kk + 1];
    }
  }

  float bestv[8];
  int   besti[8];
#pragma unroll
  for (int r = 0; r < 8; ++r) { bestv[r] = -3.0e38f; besti[r] = 0; }

  stage_codes(cn, cn2, 0, ldsC[0], ldsN[0], tid);   // prologue: chunk 0

  const int nch = BOOK / CHUNK;
#pragma unroll 1
  for (int c = 0; c < nch; ++c) {
    async_wait();
    __syncthreads();                                 // chunk c resident
    if (c + 1 < nch)                                 // prefetch chunk c+1
      stage_codes(cn, cn2, (c + 1) * CHUNK, ldsC[(c + 1) & 1],
                  ldsN[(c + 1) & 1], tid);
    const float* bufC = ldsC[c & 1];
    const float* bufN = ldsN[c & 1];

#pragma unroll 1
    for (int ct = 0; ct < CHUNK / 16; ++ct) {
      // preload all 16 B pairs, then run the uninterrupted WMMA chain
      v2f b[16];
      const float* br = bufC + (size_t)(ct * 16 + l16) * 64;
#pragma unroll
      for (int k = 0; k < 16; ++k) {
        const int kk = k * 4 + half * 2;
        b[k].x = br[kk];
        b[k].y = br[kk + 1];
      }
      v8f acc = {0.f, 0.f, 0.f, 0.f, 0.f, 0.f, 0.f, 0.f};
#pragma unroll
      for (int k = 0; k < 16; ++k) acc = wmma_f32_4(a[k], b[k], acc);

      const float c2 = bufN[ct * 16 + l16];
      const int code = c * CHUNK + ct * 16 + l16;
#pragma unroll
      for (int r = 0; r < 8; ++r) {
        const float s = 2.0f * acc[r] - c2;   // maximize == minimize dist
        if (s > bestv[r]) { bestv[r] = s; besti[r] = code; }
      }
    }
    __syncthreads();                                 // done reading buf c
  }

  // Cross-lane argmax within each 16-lane half (lower index wins ties)
#pragma unroll
  for (int r = 0; r < 8; ++r) {
    float v = bestv[r];
    int   ix = besti[r];
#pragma unroll
    for (int m = 8; m >= 1; m >>= 1) {
      const float ov = __shfl_xor(v, m, 32);
      const int   oi = __shfl_xor(ix, m, 32);
      if (ov > v || (ov == v && oi < ix)) { v = ov; ix = oi; }
    }
    if (l16 == 0) idx_out[rowbase + r + half * 8] = ix;
  }
}

// ---------------------------------------------------------------------------
// Build Xt (2048 x 1024): Xt[2b][e]=cls, Xt[2b+1][e]=codebook[idx]
// ---------------------------------------------------------------------------
__global__ __launch_bounds__(256) void build_fuse_kernel(
    const float* __restrict__ tokens, const float* __restrict__ codebook,
    const int* __restrict__ idx, float* __restrict__ Xt) {
  const int i = blockIdx.x * 256 + threadIdx.x;   // 0 .. 1M-1
  const int b = i >> 10, e = i & 1023;
  const float cls = tokens[(size_t)b * 2048 + e];
  const int id = idx[i >> 6];
  const float q = codebook[(size_t)id * 64 + (i & 63)];
  Xt[(size_t)(2 * b) * 1024 + e]     = cls;
  Xt[(size_t)(2 * b + 1) * 1024 + e] = q;
}

// ---------------------------------------------------------------------------
// Tiled GEMM: C = A(MxK, rm) @ B(NxK, rm)^T, f32 WMMA, 128x64 block / WG.
// MODE 0: store O[n*M+m] (projections, transposed layout)
// MODE 1: bias + QuickGELU, store O[m*N+n]  (final FFN)
// ---------------------------------------------------------------------------
template <int MODE>
__global__ __launch_bounds__(256) void gemm_nt_kernel(
    const float* __restrict__ A, const float* __restrict__ B,
    float* __restrict__ O, const float* __restrict__ bias,
    int M, int N, int K) {
  __shared__ float lA[128 * 64];
  __shared__ float lB[64 * 64];
  const int tid  = threadIdx.x;
  const int wave = tid >> 5;
  const int lane = tid & 31;
  const int l16  = lane & 15;
  const int half = lane >> 4;
  const int mbase = blockIdx.x * 128;
  const int nbase = blockIdx.y * 64;

  v8f acc[4];
#pragma unroll
  for (int t = 0; t < 4; ++t) acc[t] = (v8f){0.f,0.f,0.f,0.f,0.f,0.f,0.f,0.f};

#pragma unroll 1
  for (int kb = 0; kb < K; kb += 64) {
    __syncthreads();
#pragma unroll
    for (int i = tid; i < 2048; i += 256) {           // stage A: 128x64
      const int row = i >> 4, kk = (i & 15) << 2;
      copy_f4_to_lds((const float4*)(A + (size_t)(mbase + row) * K + kb + kk),
                     ((float4*)lA) + i);
    }
#pragma unroll
    for (int i = tid; i < 1024; i += 256) {           // stage B: 64x64
      const int row = i >> 4, kk = (i & 15) << 2;
      copy_f4_to_lds((const float4*)(B + (size_t)(nbase + row) * K + kb + kk),
                     ((float4*)lB) + i);
    }
    async_wait();
    __syncthreads();

    v2f a[16];
    const float* ar = lA + (size_t)(wave * 16 + l16) * 64;
#pragma unroll
    for (int k = 0; k < 16; ++k) {
      const int kk = k * 4 + half * 2;
      a[k].x = ar[kk];
      a[k].y = ar[kk + 1];
    }
#pragma unroll
    for (int nt = 0; nt < 4; ++nt) {
      v2f b[16];
      const float* br = lB + (size_t)(nt * 16 + l16) * 64;
#pragma unroll
      for (int k = 0; k < 16; ++k) {
        const int kk = k * 4 + half * 2;
        b[k].x = br[kk];
        b[k].y = br[kk + 1];
      }
#pragma unroll
      for (int k = 0; k < 16; ++k) acc[nt] = wmma_f32_4(a[k], b[k], acc[nt]);
    }
  }

  const int m0 = mbase + wave * 16;
#pragma unroll
  for (int nt = 0; nt < 4; ++nt) {
    const int n = nbase + nt * 16 + l16;
#pragma unroll
    for (int r = 0; r < 8; ++r) {
      const int m = m0 + r + half * 8;
      float v = acc[nt][r];
      if (MODE == 0) {
        O[(size_t)n * M + m] = v;
      } else {
        v += bias[n];
        v = v / (1.0f + __expf(-1.702f * v));         // QuickGELU
        O[(size_t)m * N + n] = v;
      }
    }
  }
}

// ---------------------------------------------------------------------------
// Per-batch 2x2 attention + softmax(axis=t) + Flat assembly
// Flat[b, s*1024 + d] = sum_t Vt[d,t] * A[t,s]
// ---------------------------------------------------------------------------
__global__ __launch_bounds__(256) void attn_kernel(
    const float* __restrict__ Kp, const float* __restrict__ Qp,
    const float* __restrict__ Vp, float* __restrict__ Flat) {
  const int b = blockIdx.x;
  const int tid = threadIdx.x, wave = tid >> 5, lane = tid & 31;
  const float* k0 = Kp + (size_t)(2 * b) * 1024;
  const float* k1 = k0 + 1024;
  const float* q0 = Qp + (size_t)(2 * b) * 1024;
  const float* q1 = q0 + 1024;

  float s00 = 0.f, s01 = 0.f, s10 = 0.f, s11 = 0.f;
  for (int d = tid; d < 1024; d += 256) {
    const float a0 = k0[d], a1 = k1[d], c0 = q0[d], c1 = q1[d];
    s00 += a0 * c0; s01 += a0 * c1; s10 += a1 * c0; s11 += a1 * c1;
  }
#pragma unroll
  for (int m = 16; m >= 1; m >>= 1) {
    s00 += __shfl_xor(s00, m, 32);
    s01 += __shfl_xor(s01, m, 32);
    s10 += __shfl_xor(s10, m, 32);
    s11 += __shfl_xor(s11, m, 32);
  }
  __shared__ float red[4][8];
  __shared__ float Am[4];
  if (lane == 0) { red[0][wave] = s00; red[1][wave] = s01;
                   red[2][wave] = s10; red[3][wave] = s11; }
  __syncthreads();
  if (tid == 0) {
    float t00 = 0.f, t01 = 0.f, t10 = 0.f, t11 = 0.f;
    for (int w = 0; w < 8; ++w) {
      t00 += red[0][w]; t01 += red[1][w]; t10 += red[2][w]; t11 += red[3][w];
    }
    const float sc = 0.70710678118654752f;   // 1/sqrt(2)
    t00 *= sc; t01 *= sc; t10 *= sc; t11 *= sc;
    // softmax over t (rows) per column s
    float m0 = fmaxf(t00, t10), m1 = fmaxf(t01, t11);
    float e00 = __expf(t00 - m0), e10 = __expf(t10 - m0);
    float e01 = __expf(t01 - m1), e11 = __expf(t11 - m1);
    Am[0] = e00 / (e00 + e10);   // A[t=0,s=0]
    Am[2] = e10 / (e00 + e10);   // A[t=1,s=0]
    Am[1] = e01 / (e01 + e11);   // A[t=0,s=1]
    Am[3] = e11 / (e01 + e11);   // A[t=1,s=1]
  }
  __syncthreads();
  const float A00 = Am[0], A01 = Am[1], A10 = Am[2], A11 = Am[3];
  const float* v0 = Vp + (size_t)(2 * b) * 1024;
  const float* v1 = v0 + 1024;
  float* f = Flat + (size_t)b * 2048;
  for (int d = tid; d < 1024; d += 256) {
    const float x0 = v0[d], x1 = v1[d];
    f[d]        = x0 * A00 + x1 * A10;   // s = 0
    f[1024 + d] = x0 * A01 + x1 * A11;   // s = 1
  }
}

// ---------------------------------------------------------------------------
extern "C" void kernel_launch(void* const* d_in, const int* in_sizes, int n_in,
                              void* d_out, int out_size, void* d_ws, size_t ws_size,
                              hipStream_t stream) {
  const float* tokens   = (const float*)d_in[0];   // (1024,2,1024)
  const float* codebook = (const float*)d_in[1];   // (16384,64)
  const float* Kw       = (const float*)d_in[2];   // (1024,1024)
  const float* Qw       = (const float*)d_in[3];
  const float* Vw       = (const float*)d_in[4];
  const float* Wf       = (const float*)d_in[5];   // (1024,2048)
  const float* bf       = (const float*)d_in[6];   // (1024,)
  float* out = (float*)d_out;                      // (1024,1,1024)

  // workspace layout (floats)
  float* w    = (float*)d_ws;
  float* tn   = w;                         // 1M
  float* cn   = tn + (size_t)NTOK * 64;    // 1M
  float* cn2  = cn + (size_t)BOOK * 64;    // 16K
  int*   idx  = (int*)(cn2 + BOOK);        // 16K
  float* Xt   = (float*)(idx + NTOK);      // 2M
  float* Kp   = Xt + (size_t)2 * BS * DIM; // 2M
  float* Qp   = Kp + (size_t)2 * BS * DIM; // 2M
  float* Vp   = Qp + (size_t)2 * BS * DIM; // 2M
  float* Flat = Vp + (size_t)2 * BS * DIM; // 2M

  norm_tokens_kernel<<<NTOK / 8, 256, 0, stream>>>(tokens, tn);
  norm_codebook_kernel<<<BOOK / 8, 256, 0, stream>>>(codebook, cn, cn2);
  nn_search_kernel<<<NTOK / 128, 256, 0, stream>>>(tn, cn, cn2, idx);
  build_fuse_kernel<<<(BS * DIM) / 256, 256, 0, stream>>>(tokens, codebook, idx, Xt);

  // Projections: Proj^T[col][d] = sum_e W[d,e] * Xt[col,e]
  gemm_nt_kernel<0><<<dim3(DIM / 128, (2 * BS) / 64), 256, 0, stream>>>(
      Kw, Xt, Kp, nullptr, DIM, 2 * BS, DIM);
  gemm_nt_kernel<0><<<dim3(DIM / 128, (2 * BS) / 64), 256, 0, stream>>>(
      Qw, Xt, Qp, nullptr, DIM, 2 * BS, DIM);
  gemm_nt_kernel<0><<<dim3(DIM / 128, (2 * BS) / 64), 256, 0, stream>>>(
      Vw, Xt, Vp, nullptr, DIM, 2 * BS, DIM);

  attn_kernel<<<BS, 256, 0, stream>>>(Kp, Qp, Vp, Flat);

  // h = QuickGELU(Flat @ Wf^T + bf), written row-major (1024 x 1024)
  gemm_nt_kernel<1><<<dim3(BS / 128, DIM / 64), 256, 0, stream>>>(
      Flat, Wf, out, bf, BS, DIM, 2 * DIM);
}